// CrossModalAttentionModule_43052752175502
// MI455X (gfx1250) — compile-verified
//
#include <hip/hip_runtime.h>
#include <hip/hip_bf16.h>
#include <stdint.h>

// CDNA5 (gfx1250, wave32) implementation.
// Math restructuring: all large projections are pulled through the final
// contraction over C so the only big compute is one batched GEMM
//   T[b] = G[2S x C] x feat[b][C x N]   (G = stacked QK / VV matrices)
// done with V_WMMA_F32_16X16X4_F32 (full fp32 accuracy). Global->LDS staging
// uses double-buffered GLOBAL_LOAD_ASYNC_TO_LDS_B128 (ASYNCcnt) so loads for
// tile t+1 overlap WMMA compute on tile t. Edge tiles are handled by index
// clamping (OOB rows/cols only feed never-stored accumulator elements).

typedef float v2f __attribute__((ext_vector_type(2)));
typedef float v8f __attribute__((ext_vector_type(8)));

#define TM 64
#define TN 32
#define KT 32
#define APAD 4   // row stride 36 dwords: 16B-aligned row starts, conflict-free frags

// low 32 bits of a flat LDS pointer == wave-relative LDS byte offset (ISA 10.2)
__device__ __forceinline__ uint32_t lds_off(const void* p) {
    return (uint32_t)(uintptr_t)p;
}
// 16B global -> LDS async copy (per-lane), tracked with ASYNCcnt
__device__ __forceinline__ void async_copy16(uint32_t lds, const float* g) {
    asm volatile("global_load_async_to_lds_b128 %0, %1, off"
                 :: "v"(lds), "v"((unsigned long long)(uintptr_t)g)
                 : "memory");
}
// non-temporal variant for single-use streams (feat)
__device__ __forceinline__ void async_copy16_nt(uint32_t lds, const float* g) {
    asm volatile("global_load_async_to_lds_b128 %0, %1, off th:TH_LOAD_NT"
                 :: "v"(lds), "v"((unsigned long long)(uintptr_t)g)
                 : "memory");
}
__device__ __forceinline__ void async_wait0() {
    asm volatile("s_wait_asynccnt 0" ::: "memory");
}
__device__ __forceinline__ void async_wait3() {  // keep newest 3 in flight
    asm volatile("s_wait_asynccnt 3" ::: "memory");
}

// ---------------------------------------------------------------------------
// Generic fp32 WMMA GEMM:  Out[r,n] = sum_k A[r,k] * B(k,n)  (+ bias[n])
//  - B(k,n) = transB ? Bt[n*ldb + k] : B[k*ldb + n]
//  - batched via blockIdx.z with element strides (0 to broadcast)
// Block = 256 threads = 8 waves; 64x32 block tile; K staged in steps of 32.
// Fast (async, double-buffered) path requires: !transB, K % 32 == 0,
//   lda % 4 == 0, ldb % 4 == 0, Nn % 4 == 0, 16B-aligned A/B bases
// (guaranteed by all call sites that use it).
// ---------------------------------------------------------------------------
__global__ __launch_bounds__(256)
void gemm_wmma_f32(const float* __restrict__ A, int lda, long long strideA,
                   const float* __restrict__ B, int ldb, long long strideB,
                   int transB, int ntB,
                   const float* __restrict__ bias,
                   float* __restrict__ Out, int ldo, long long strideO,
                   int R, int K, int Nn)
{
    const long long bz = blockIdx.z;
    A   += bz * strideA;
    B   += bz * strideB;
    Out += bz * strideO;

    __shared__ float As[2][TM][KT + APAD];   // [buf][row][k]
    __shared__ float Bs[2][KT][TN + APAD];   // [buf][k][col]

    const int tid  = threadIdx.x;
    const int lane = tid & 31;
    const int wave = tid >> 5;
    const int wrow = wave & 3;    // 0..3 -> 16-row band
    const int wcol = wave >> 2;   // 0..1 -> 16-col band

    const int rowBase = blockIdx.y * TM;
    const int colBase = blockIdx.x * TN;

    v8f acc = {0.f, 0.f, 0.f, 0.f, 0.f, 0.f, 0.f, 0.f};

    // fp32 WMMA fragment coords (ISA 7.12.2, 32-bit A 16x4):
    //   lane L: row/col = L%16 ; K pair base = (L/16)*2
    const int fr    = lane & 15;
    const int kHalf = (lane >> 4) << 1;
    const int arow  = wrow * 16 + fr;
    const int bcol  = wcol * 16 + fr;

    const bool fastAll = (!transB) && (K % KT == 0);

    if (fastAll) {
        // per-thread staging coords (fixed across K):
        //   A: two chunks (r0, r0+32) x 4 floats ; B: row r0 x 4 floats
        const int r0  = tid >> 3;            // 0..31
        const int c4  = (tid & 7) << 2;      // 0,4,..,28
        // clamp OOB rows/cols to valid addresses: those lanes only feed
        // accumulator elements that the store phase discards.
        int gr0 = rowBase + r0;        if (gr0 > R - 1) gr0 = R - 1;
        int gr1 = rowBase + r0 + 32;   if (gr1 > R - 1) gr1 = R - 1;
        int gc  = colBase + c4;        if (gc > Nn - 4) gc = Nn - 4;

        const float* aSrc0 = A + (long long)gr0 * lda + c4;
        const float* aSrc1 = A + (long long)gr1 * lda + c4;
        const float* bSrc  = B + (long long)r0  * ldb + gc;

        const uint32_t aL0 = lds_off(&As[0][r0][c4]);
        const uint32_t aL1 = lds_off(&As[0][r0 + 32][c4]);
        const uint32_t bL  = lds_off(&Bs[0][r0][c4]);
        const uint32_t sA  = (uint32_t)(sizeof(float) * TM * (KT + APAD));
        const uint32_t sB  = (uint32_t)(sizeof(float) * KT * (TN + APAD));

        const int nk = K / KT;
        // prologue: issue tile 0 into buffer 0
        async_copy16(aL0, aSrc0);
        async_copy16(aL1, aSrc1);
        if (ntB) async_copy16_nt(bL, bSrc); else async_copy16(bL, bSrc);

        for (int t = 0; t < nk; ++t) {
            if (t + 1 < nk) {
                const uint32_t buf = (uint32_t)((t + 1) & 1);
                const long long ko = (long long)(t + 1) * KT;
                async_copy16(aL0 + buf * sA, aSrc0 + ko);
                async_copy16(aL1 + buf * sA, aSrc1 + ko);
                if (ntB) async_copy16_nt(bL + buf * sB, bSrc + ko * ldb);
                else     async_copy16   (bL + buf * sB, bSrc + ko * ldb);
                async_wait3();   // tile t's 3 loads done; t+1's may fly
            } else {
                async_wait0();
            }
            __syncthreads();
            {
                const float (*Ab)[KT + APAD] = As[t & 1];
                const float (*Bb)[TN + APAD] = Bs[t & 1];
                #pragma unroll
                for (int kk = 0; kk < KT; kk += 4) {
                    v2f a, b;
                    a.x = Ab[arow][kk + kHalf];
                    a.y = Ab[arow][kk + kHalf + 1];
                    b.x = Bb[kk + kHalf][bcol];
                    b.y = Bb[kk + kHalf + 1][bcol];
                    acc = __builtin_amdgcn_wmma_f32_16x16x4_f32(
                        false, a, false, b, (short)0, acc, false, false);
                }
            }
            __syncthreads();  // protect buffer reuse at t+2's issue
        }
    } else {
        // ---- guarded scalar path (transB / K-tail); small GEMMs only
        for (int k0 = 0; k0 < K; k0 += KT) {
            for (int i = tid; i < TM * KT; i += 256) {
                int r = i / KT, kk = i - r * KT;
                int gr = rowBase + r, gk = k0 + kk;
                float v = 0.f;
                if (gr < R && gk < K) v = A[(long long)gr * lda + gk];
                As[0][r][kk] = v;
            }
            if (!transB) {
                for (int i = tid; i < TN * KT; i += 256) {
                    int kk = i / TN, n = i - kk * TN;
                    int gk = k0 + kk, gn = colBase + n;
                    float v = 0.f;
                    if (gk < K && gn < Nn) v = B[(long long)gk * ldb + gn];
                    Bs[0][kk][n] = v;
                }
            } else {
                for (int i = tid; i < TN * KT; i += 256) {
                    int n = i / KT, kk = i - n * KT;
                    int gk = k0 + kk, gn = colBase + n;
                    float v = 0.f;
                    if (gk < K && gn < Nn) v = B[(long long)gn * ldb + gk];
                    Bs[0][kk][n] = v;
                }
            }
            __syncthreads();
            #pragma unroll
            for (int kk = 0; kk < KT; kk += 4) {
                v2f a, b;
                a.x = As[0][arow][kk + kHalf];
                a.y = As[0][arow][kk + kHalf + 1];
                b.x = Bs[0][kk + kHalf][bcol];
                b.y = Bs[0][kk + kHalf + 1][bcol];
                acc = __builtin_amdgcn_wmma_f32_16x16x4_f32(
                    false, a, false, b, (short)0, acc, false, false);
            }
            __syncthreads();
        }
    }

    // C/D layout: VGPR r -> row r (lanes 0-15) / row r+8 (lanes 16-31), col = lane%16
    const int outCol = colBase + wcol * 16 + (lane & 15);
    const int rBase  = rowBase + wrow * 16 + ((lane >> 4) ? 8 : 0);
    if (outCol < Nn) {
        float bval = bias ? bias[outCol] : 0.f;
        #pragma unroll
        for (int r = 0; r < 8; ++r) {
            int outRow = rBase + r;
            if (outRow < R) Out[(long long)outRow * ldo + outCol] = acc[r] + bval;
        }
    }
}

// ---------------------------------------------------------------------------
// wave32 reductions
// ---------------------------------------------------------------------------
__device__ __forceinline__ float wave_sum(float v) {
    #pragma unroll
    for (int off = 16; off > 0; off >>= 1) v += __shfl_xor(v, off, 32);
    return v;
}
__device__ __forceinline__ float wave_max(float v) {
    #pragma unroll
    for (int off = 16; off > 0; off >>= 1) v = fmaxf(v, __shfl_xor(v, off, 32));
    return v;
}

// feat_pool[b,c] = mean_n feat[b,c,n] ; one wave per (b,c) row
__global__ __launch_bounds__(256)
void pool_kernel(const float* __restrict__ feat, float* __restrict__ pool,
                 int rows, int N)
{
    int row  = (blockIdx.x * blockDim.x + threadIdx.x) >> 5;
    int lane = threadIdx.x & 31;
    if (row >= rows) return;
    const float* p = feat + (long long)row * N;
    float s = 0.f;
    for (int n = lane; n < N; n += 32) s += p[n];
    s = wave_sum(s);
    if (lane == 0) pool[row] = s * (1.0f / (float)N);
}

// wbv[s] = sum_m WoV[s,m]*bv[m] ; boV[s] = sum_c V_final[s,c]*bo[c]
__global__ __launch_bounds__(256)
void scalars_kernel(const float* __restrict__ WoV, const float* __restrict__ bv,
                    const float* __restrict__ Vf,  const float* __restrict__ bo,
                    float* __restrict__ wbv, float* __restrict__ boV,
                    int S, int M, int C)
{
    int s    = (blockIdx.x * blockDim.x + threadIdx.x) >> 5;
    int lane = threadIdx.x & 31;
    if (s >= S) return;
    float a = 0.f, b = 0.f;
    const float* wr = WoV + (long long)s * M;
    const float* vr = Vf  + (long long)s * C;
    for (int m = lane; m < M; m += 32) a += wr[m] * bv[m];
    for (int c = lane; c < C; c += 32) b += vr[c] * bo[c];
    a = wave_sum(a);
    b = wave_sum(b);
    if (lane == 0) { wbv[s] = a; boV[s] = b; }
}

// softmax over N of T0 row, dot with T1 row, combine. One wave per (b,s).
__global__ __launch_bounds__(256)
void final_kernel(const float* __restrict__ T, const float* __restrict__ fpV,
                  const float* __restrict__ wbv, const float* __restrict__ boV,
                  float* __restrict__ out, int B, int S, int N, float ratio)
{
    int row  = (blockIdx.x * blockDim.x + threadIdx.x) >> 5;  // = b*S + s
    int lane = threadIdx.x & 31;
    if (row >= B * S) return;
    int b = row / S;
    int s = row - b * S;
    const float* P  = T + ((long long)b * 2 * S + s) * N;       // scores (qbk shift dropped: softmax-invariant)
    const float* V2 = T + ((long long)b * 2 * S + S + s) * N;   // vW (pre-bias)

    float mx = -3.402823466e38f;
    for (int n = lane; n < N; n += 32) mx = fmaxf(mx, P[n]);
    mx = wave_max(mx);

    float den = 0.f, num = 0.f;
    for (int n = lane; n < N; n += 32) {
        float e = __expf(P[n] - mx);
        den += e;
        num += e * V2[n];
    }
    den = wave_sum(den);
    num = wave_sum(num);
    if (lane == 0)
        out[row] = fpV[row] + ratio * (num / den + wbv[s] + boV[s]);
}

// ---------------------------------------------------------------------------
extern "C" void kernel_launch(void* const* d_in, const int* in_sizes, int n_in,
                              void* d_out, int out_size, void* d_ws, size_t ws_size,
                              hipStream_t stream) {
    (void)in_sizes; (void)n_in; (void)out_size; (void)ws_size;

    const int Bn = 128, Cn = 2048, Nn = 196, Sn = 312, Ln = 300, Mn = 1024;
    const float RATIO = 1.0f;

    const float* feat    = (const float*)d_in[0];   // [B,C,N]
    const float* att_emb = (const float*)d_in[1];   // [S,L]
    const float* Wq      = (const float*)d_in[2];   // [M,L]
    const float* bq      = (const float*)d_in[3];   // [M]
    const float* Wk      = (const float*)d_in[4];   // [M,C]
    // d_in[5] = bk: softmax shift-invariant -> provably no effect on output
    const float* Wv      = (const float*)d_in[6];   // [M,C]
    const float* bv      = (const float*)d_in[7];   // [M]
    const float* Wo      = (const float*)d_in[8];   // [C,M]
    const float* bo      = (const float*)d_in[9];   // [C]
    const float* Vf      = (const float*)d_in[10];  // [S,C]
    float* out = (float*)d_out;                     // [B,S]

    // workspace layout (floats), total ~71.5 MB
    float* ws   = (float*)d_ws;
    float* wsQ   = ws;                         // [S,M]  query
    float* wsWoV = wsQ   + (size_t)Sn * Mn;    // [S,M]  V_final @ Wo
    float* wsG   = wsWoV + (size_t)Sn * Mn;    // [2S,C] rows 0..S-1 = QK, S..2S-1 = VV
    float* wsPool= wsG   + (size_t)2 * Sn * Cn;// [B,C]
    float* wsFpV = wsPool+ (size_t)Bn * Cn;    // [B,S]
    float* wsWbv = wsFpV + (size_t)Bn * Sn;    // [S]
    float* wsBoV = wsWbv + Sn;                 // [S]
    float* wsT   = wsBoV + Sn + 40;            // [B][2S][N] (offset keeps 16B align)

    dim3 blk(256);
    #define GEMM_GRID(RR, NN, ZZ) dim3(((NN) + TN - 1) / TN, ((RR) + TM - 1) / TM, (ZZ))

    // 1) query = att_emb @ Wq^T + bq          [S,M]   (transB -> guarded path)
    gemm_wmma_f32<<<GEMM_GRID(Sn, Mn, 1), blk, 0, stream>>>(
        att_emb, Ln, 0, Wq, Ln, 0, /*transB=*/1, 0, bq, wsQ, Mn, 0, Sn, Ln, Mn);

    // 2) WoV = V_final @ Wo                    [S,M]
    gemm_wmma_f32<<<GEMM_GRID(Sn, Mn, 1), blk, 0, stream>>>(
        Vf, Cn, 0, Wo, Mn, 0, 0, 0, nullptr, wsWoV, Mn, 0, Sn, Cn, Mn);

    // 3) QK = query @ Wk                       [S,C] -> G rows [0,S)
    gemm_wmma_f32<<<GEMM_GRID(Sn, Cn, 1), blk, 0, stream>>>(
        wsQ, Mn, 0, Wk, Cn, 0, 0, 0, nullptr, wsG, Cn, 0, Sn, Mn, Cn);

    // 4) VV = WoV @ Wv                         [S,C] -> G rows [S,2S)
    gemm_wmma_f32<<<GEMM_GRID(Sn, Cn, 1), blk, 0, stream>>>(
        wsWoV, Mn, 0, Wv, Cn, 0, 0, 0, nullptr, wsG + (size_t)Sn * Cn, Cn, 0, Sn, Mn, Cn);

    // 5) wbv[s], boV[s]
    {
        int nblk = (Sn * 32 + 255) / 256;
        scalars_kernel<<<nblk, blk, 0, stream>>>(wsWoV, bv, Vf, bo, wsWbv, wsBoV, Sn, Mn, Cn);
    }

    // 6) feat_pool[b,c] = mean_n feat
    {
        int rows = Bn * Cn;
        int nblk = (rows * 32 + 255) / 256;
        pool_kernel<<<nblk, blk, 0, stream>>>(feat, wsPool, rows, Nn);
    }

    // 7) fpV = feat_pool @ V_final^T           [B,S]   (transB -> guarded path)
    gemm_wmma_f32<<<GEMM_GRID(Bn, Sn, 1), blk, 0, stream>>>(
        wsPool, Cn, 0, Vf, Cn, 0, /*transB=*/1, 0, nullptr, wsFpV, Sn, 0, Bn, Cn, Sn);

    // 8) batched: T[b] = G[2S,C] @ feat[b][C,N]  -> [B][2S][N]   (the big one)
    //    feat is a single-use 205MB stream -> NT hint keeps G resident in L2
    gemm_wmma_f32<<<GEMM_GRID(2 * Sn, Nn, Bn), blk, 0, stream>>>(
        wsG, Cn, 0,
        feat, Nn, (long long)Cn * Nn, 0, /*ntB=*/1, nullptr,
        wsT, Nn, (long long)2 * Sn * Nn,
        2 * Sn, Cn, Nn);

    // 9) softmax over N + weighted dot + combine -> out[B,S]
    {
        int rows = Bn * Sn;
        int nblk = (rows * 32 + 255) / 256;
        final_kernel<<<nblk, blk, 0, stream>>>(wsT, wsFpV, wsWbv, wsBoV, out, Bn, Sn, Nn, RATIO);
    }
    #undef GEMM_GRID
}